// LMKAN2DLayer_89910845374936
// MI455X (gfx1250) — compile-verified
//
#include <hip/hip_runtime.h>
#include <stdint.h>

// Problem constants (match reference)
#define GGRID 32
#define NB 33                 // GGRID+1 grid nodes per axis
#define ODIM 256
#define PDIM 64
#define BATCH 4096
#define RROWS (NB * NB * ODIM)          // 278784 "rows" of 64 (p) in source layout
#define FPT_FLOATS ((size_t)PDIM * NB * NB * ODIM)   // 17,842,176
#define FPT_BYTES  (FPT_FLOATS * sizeof(float))      // 71,368,704
#define TAB_STRIDE 1280                  // per-batch: 64*int (256B) + 64*float4 (1024B)

typedef float vf4 __attribute__((ext_vector_type(4)));  // native vector for NT loads

// ---- CDNA5 async global->LDS path (builtin existence proven by round-1 diag)
#if defined(__has_builtin)
#if __has_builtin(__builtin_amdgcn_global_load_async_to_lds_b128)
#define HAVE_ASYNC_LDS 1
#endif
#if __has_builtin(__builtin_amdgcn_s_wait_asynccnt)
#define HAVE_WAIT_ASYNC_BUILTIN 1
#endif
#endif

#if defined(HAVE_ASYNC_LDS)
#define AS1 __attribute__((address_space(1)))
#define AS3 __attribute__((address_space(3)))
typedef int vi4 __attribute__((vector_size(16)));
__device__ __forceinline__ void async_copy_b128(const void* g, void* l) {
  __builtin_amdgcn_global_load_async_to_lds_b128((AS1 vi4*)g, (AS3 vi4*)l, 0, 0);
}
__device__ __forceinline__ void wait_async0() {
#if defined(HAVE_WAIT_ASYNC_BUILTIN)
  __builtin_amdgcn_s_wait_asynccnt(0);
#else
  asm volatile("s_wait_asynccnt 0x0" ::: "memory");
#endif
  asm volatile("" ::: "memory");  // keep LDS reads below the wait
}
#endif

// ---- kernel 1: per-(b,p) bucketize + bilinear weights ----------------------
__global__ __launch_bounds__(256) void lmkan_prep(const float* __restrict__ x,
                                                  const float* __restrict__ borders,
                                                  char* __restrict__ tab) {
  __shared__ float bd[NB];
  int t = threadIdx.x;
  if (t < NB) bd[t] = borders[t];
  __syncthreads();

  int id = blockIdx.x * 256 + t;        // 0 .. BATCH*PDIM-1
  int b = id >> 6;
  int p = id & 63;

  float2 ab = *(const float2*)(x + (size_t)b * 128 + 2 * p);
  float lo = bd[0], hi = bd[GGRID];
  float a = fminf(fmaxf(ab.x, lo), hi);
  float c = fminf(fmaxf(ab.y, lo), hi);

  // searchsorted(borders, v, side='right') - 1, for v in [lo, hi]
  int i = 0, j = 0;
#pragma unroll
  for (int k = 1; k <= GGRID; ++k) {
    i += (a >= bd[k]) ? 1 : 0;
    j += (c >= bd[k]) ? 1 : 0;
  }
  i = min(i, GGRID - 1);
  j = min(j, GGRID - 1);

  float tt = (a - bd[i]) / (bd[i + 1] - bd[i]);
  float uu = (c - bd[j]) / (bd[j + 1] - bd[j]);

  // base float-offset into fpT (p, i, j, o) layout
  int cell = ((p * NB + i) * NB + j) * ODIM;

  char* tb = tab + (size_t)b * TAB_STRIDE;
  ((int*)tb)[p] = cell;
  // order: (w00, w01, w10, w11) matching corner offsets (+0, +256, +8448, +8704)
  ((float4*)(tb + 256))[p] =
      make_float4((1.f - tt) * (1.f - uu), (1.f - tt) * uu, tt * (1.f - uu), tt * uu);
}

// ---- kernel 2: transpose (i,j,o,p) -> (p,i,j,o)  [R x 64 -> 64 x R] --------
__global__ __launch_bounds__(256) void lmkan_transpose(const float* __restrict__ src,
                                                       float* __restrict__ dst) {
  __shared__ float tileT[64 * 68];      // [p][rr], stride 68 words (bank-spread)
  int t = threadIdx.x;
  size_t r0 = (size_t)blockIdx.x * 64;  // 4356 blocks cover RROWS
  const float* gsrc = src + r0 * 64;    // tile is a contiguous 16KB chunk

#pragma unroll
  for (int k = 0; k < 4; ++k) {
    int q = k * 256 + t;                // 0..1023 float4 slots
    int rr = q & 63;                    // source row within tile
    int cg = q >> 6;                    // col group (0..15)
    // src is read exactly once: non-temporal so it doesn't evict fpT from L2
    vf4 v = __builtin_nontemporal_load((const vf4*)(gsrc + rr * 64 + cg * 4));
    int c0 = cg * 4;
    tileT[(c0 + 0) * 68 + rr] = v.x;    // transposed write: lanes -> distinct banks
    tileT[(c0 + 1) * 68 + rr] = v.y;
    tileT[(c0 + 2) * 68 + rr] = v.z;
    tileT[(c0 + 3) * 68 + rr] = v.w;
  }
  __syncthreads();
#pragma unroll
  for (int k = 0; k < 4; ++k) {
    int q = k * 256 + t;
    int p = q >> 4;                     // 0..63
    int rg = q & 15;                    // row group
    float4 v;
    v.x = tileT[p * 68 + rg * 4 + 0];
    v.y = tileT[p * 68 + rg * 4 + 1];
    v.z = tileT[p * 68 + rg * 4 + 2];
    v.w = tileT[p * 68 + rg * 4 + 3];
    *(float4*)(dst + (size_t)p * RROWS + r0 + rg * 4) = v;
  }
}

// ---- kernel 3: gather + bilinear accumulate, 1 wave = 1 batch --------------
__device__ __forceinline__ void fma4(float4& a, float s, float4 v) {
  a.x = fmaf(s, v.x, a.x);
  a.y = fmaf(s, v.y, a.y);
  a.z = fmaf(s, v.z, a.z);
  a.w = fmaf(s, v.w, a.w);
}

__global__ __launch_bounds__(256) void lmkan_main(const float* __restrict__ fpT,
                                                  const char* __restrict__ tab,
                                                  float* __restrict__ out) {
  __shared__ __align__(16) char smem[8 * TAB_STRIDE];
  int lane = threadIdx.x & 31;
  int wv = threadIdx.x >> 5;
  int b = blockIdx.x * 8 + wv;

  const char* gTab = tab + (size_t)b * TAB_STRIDE;

#if defined(HAVE_ASYNC_LDS)
  char* myTab = smem + wv * TAB_STRIDE;
  {
    int off = lane * 16;
    async_copy_b128(gTab + off, myTab + off);                     // bytes [0,512)
    async_copy_b128(gTab + 512 + off, myTab + 512 + off);         // bytes [512,1024)
    if (lane < 16) async_copy_b128(gTab + 1024 + off, myTab + 1024 + off);
    wait_async0();   // wave-local staging: no barrier needed
  }
  const int* cells = (const int*)myTab;
  const float4* ws4 = (const float4*)(myTab + 256);
#else
  (void)smem;
  const int* cells = (const int*)gTab;
  const float4* ws4 = (const float4*)(gTab + 256);
#endif

  float4 acc0 = make_float4(0.f, 0.f, 0.f, 0.f);
  float4 acc1 = make_float4(0.f, 0.f, 0.f, 0.f);
  int o0 = lane * 4;  // lane covers o0..o0+3 and o0+128..o0+131

#pragma unroll 4
  for (int p = 0; p < PDIM; ++p) {
    int cell = cells[p];
    cell = __builtin_amdgcn_readfirstlane(cell);  // uniform -> scalar base addr
    float4 w = ws4[p];
    const float* base = fpT + cell;

    float4 f00a = *(const float4*)(base + o0);
    float4 f00b = *(const float4*)(base + o0 + 128);
    float4 f01a = *(const float4*)(base + 256 + o0);        // (i, j+1)
    float4 f01b = *(const float4*)(base + 256 + o0 + 128);
    float4 f10a = *(const float4*)(base + 8448 + o0);       // (i+1, j) = +33*256
    float4 f10b = *(const float4*)(base + 8448 + o0 + 128);
    float4 f11a = *(const float4*)(base + 8704 + o0);       // (i+1, j+1)
    float4 f11b = *(const float4*)(base + 8704 + o0 + 128);

    fma4(acc0, w.x, f00a); fma4(acc1, w.x, f00b);
    fma4(acc0, w.y, f01a); fma4(acc1, w.y, f01b);
    fma4(acc0, w.z, f10a); fma4(acc1, w.z, f10b);
    fma4(acc0, w.w, f11a); fma4(acc1, w.w, f11b);
  }

  float* po = out + (size_t)b * ODIM + o0;
  *(float4*)(po) = acc0;
  *(float4*)(po + 128) = acc1;
}

// ---- launcher ---------------------------------------------------------------
extern "C" void kernel_launch(void* const* d_in, const int* in_sizes, int n_in,
                              void* d_out, int out_size, void* d_ws, size_t ws_size,
                              hipStream_t stream) {
  (void)in_sizes; (void)n_in; (void)out_size; (void)ws_size;
  const float* fparam  = (const float*)d_in[0];  // (33,33,256,64)
  const float* x       = (const float*)d_in[1];  // (4096,128)
  const float* borders = (const float*)d_in[2];  // (33,)
  float* out = (float*)d_out;

  char*  ws  = (char*)d_ws;
  float* fpT = (float*)ws;                 // 71,368,704 B: (p,i,j,o) layout
  char*  tab = ws + FPT_BYTES;             // 4096 * 1280 B: per-batch cell+weight table

  lmkan_prep<<<(BATCH * PDIM) / 256, 256, 0, stream>>>(x, borders, tab);
  lmkan_transpose<<<RROWS / 64, 256, 0, stream>>>(fparam, fpT);
  lmkan_main<<<BATCH / 8, 256, 0, stream>>>(fpT, tab, out);
}